// Glove_33054068310228
// MI455X (gfx1250) — compile-verified
//
#include <hip/hip_runtime.h>

typedef float v2f __attribute__((ext_vector_type(2)));
typedef float v8f __attribute__((ext_vector_type(8)));

#if defined(__has_builtin)
# if __has_builtin(__builtin_amdgcn_wmma_f32_16x16x4_f32)
#  define HAVE_WMMA_F32X4 1
# endif
#endif
#ifndef HAVE_WMMA_F32X4
# define HAVE_WMMA_F32X4 0
#endif

#define XMAX_INV 0.2f   // 1/5.0
#define ALPHA    0.75f

// One wave handles tiles of 16 pairs. Dot products computed as the diagonal of
// L(16x64) x R^T(64x16) via chained V_WMMA_F32_16X16X4_F32.
__global__ __launch_bounds__(256) void glove_loss_kernel(
    const float* __restrict__ l_vecs, const float* __restrict__ r_vecs,
    const float* __restrict__ l_bias, const float* __restrict__ r_bias,
    const int* __restrict__ iidx, const int* __restrict__ jidx,
    const float* __restrict__ count, float* __restrict__ partials,
    int numTiles)
{
    const int lane         = threadIdx.x & 31;
    const int waveInBlock  = threadIdx.x >> 5;
    const int wavesPerBlk  = blockDim.x >> 5;
    const int globalWave   = blockIdx.x * wavesPerBlk + waveInBlock;
    const int totalWaves   = gridDim.x * wavesPerBlk;

    const int half = lane >> 4;   // K sub-pair selector for A/B fragments
    const int prow = lane & 15;   // row (A) / column (B) this lane feeds

    float acc_local = 0.0f;

    for (int tile = globalWave; tile < numTiles; tile += totalWaves) {
        const long base = (long)tile * 16;

        // Row indices for the fragment rows this lane loads (i/j streamed once -> NT)
        const int li = __builtin_nontemporal_load(&iidx[base + prow]);
        const int lj = __builtin_nontemporal_load(&jidx[base + prow]);
        const float* lrow = l_vecs + (size_t)li * 64 + 2 * half;
        const float* rrow = r_vecs + (size_t)lj * 64 + 2 * half;

        float dot;
        int   p;
        bool  active;
#if HAVE_WMMA_F32X4
        v8f acc = {};
        #pragma unroll
        for (int k = 0; k < 64; k += 4) {
            // A fragment: lane = M = prow, elements K = k+2*half .. +1  (b64 gather)
            // B fragment: lane = N = prow (r-row), same K mapping      (b64 gather)
            v2f a = *(const v2f*)(lrow + k);
            v2f b = *(const v2f*)(rrow + k);
            acc = __builtin_amdgcn_wmma_f32_16x16x4_f32(
                false, a, false, b, (short)0, acc, false, false);
        }
        // Diagonal extraction from the 16x16 f32 C layout:
        //  pair d in [0,7]  -> lane d,    acc[d]
        //  pair d in [8,15] -> lane d+16, acc[d-8]
        const int t = lane & 7;
        float sel = acc[0];
        #pragma unroll
        for (int q = 1; q < 8; ++q) sel = (t == q) ? acc[q] : sel;
        dot    = sel;
        active = (lane < 8) || (lane >= 24);
        p      = t + ((lane & 16) ? 8 : 0);
#else
        // VALU fallback: lanes L and L+16 jointly accumulate pair prow's dot.
        float partial = 0.0f;
        #pragma unroll
        for (int k = 0; k < 64; k += 4) {
            v2f a = *(const v2f*)(lrow + k);
            v2f b = *(const v2f*)(rrow + k);
            partial = fmaf(a.x, b.x, partial);
            partial = fmaf(a.y, b.y, partial);
        }
        partial += __shfl_xor(partial, 16, 32);
        dot    = partial;
        active = (lane < 16);
        p      = prow;
#endif
        float contrib = 0.0f;
        if (active) {
            const int   ii = iidx[base + p];
            const int   jj = jidx[base + p];
            const float c  = __builtin_nontemporal_load(&count[base + p]);
            const float pred = dot + l_bias[ii] + r_bias[jj] - __builtin_logf(c);
            const float w = (c < 5.0f)
                          ? __builtin_exp2f(ALPHA * __builtin_log2f(c * XMAX_INV))
                          : 1.0f;
            contrib = pred * pred * w;
        }
        acc_local += contrib;
    }

    // wave reduce (wave32)
    float s = acc_local;
    s += __shfl_xor(s, 16, 32);
    s += __shfl_xor(s,  8, 32);
    s += __shfl_xor(s,  4, 32);
    s += __shfl_xor(s,  2, 32);
    s += __shfl_xor(s,  1, 32);

    __shared__ float sdata[32];
    if (lane == 0) sdata[waveInBlock] = s;
    __syncthreads();
    if (threadIdx.x == 0) {
        float bsum = 0.0f;
        for (int w = 0; w < wavesPerBlk; ++w) bsum += sdata[w];
        partials[blockIdx.x] = bsum;   // exactly one write per block (d_ws)
    }
}

// Deterministic second-stage reduction of per-block partials -> scalar output.
__global__ __launch_bounds__(256) void glove_reduce_kernel(
    const float* __restrict__ partials, int n, float* __restrict__ out)
{
    float s = 0.0f;
    for (int idx = threadIdx.x; idx < n; idx += blockDim.x)
        s += partials[idx];
    s += __shfl_xor(s, 16, 32);
    s += __shfl_xor(s,  8, 32);
    s += __shfl_xor(s,  4, 32);
    s += __shfl_xor(s,  2, 32);
    s += __shfl_xor(s,  1, 32);

    __shared__ float sm[8];
    const int lane = threadIdx.x & 31;
    const int wv   = threadIdx.x >> 5;
    if (lane == 0) sm[wv] = s;
    __syncthreads();
    if (threadIdx.x == 0) {
        float t = 0.0f;
        const int nw = blockDim.x >> 5;
        for (int q = 0; q < nw; ++q) t += sm[q];
        out[0] = t;
    }
}

extern "C" void kernel_launch(void* const* d_in, const int* in_sizes, int n_in,
                              void* d_out, int out_size, void* d_ws, size_t ws_size,
                              hipStream_t stream) {
    const float* l_vecs = (const float*)d_in[0];
    const float* r_vecs = (const float*)d_in[1];
    const float* l_bias = (const float*)d_in[2];
    const float* r_bias = (const float*)d_in[3];
    const int*   iidx   = (const int*)d_in[4];
    const int*   jidx   = (const int*)d_in[5];
    const float* count  = (const float*)d_in[6];
    float* out      = (float*)d_out;
    float* partials = (float*)d_ws;

    const int B        = in_sizes[4];
    const int numTiles = B >> 4;              // B = 1048576 -> 65536 tiles of 16 pairs

    int blocks = 1024;                        // 8192 waves -> 8 tiles/wave
    const int maxBlocksByWs = (int)(ws_size / sizeof(float));
    if (maxBlocksByWs > 0 && blocks > maxBlocksByWs) blocks = maxBlocksByWs;
    if (blocks < 1) blocks = 1;

    glove_loss_kernel<<<blocks, 256, 0, stream>>>(
        l_vecs, r_vecs, l_bias, r_bias, iidx, jidx, count, partials, numTiles);
    glove_reduce_kernel<<<1, 256, 0, stream>>>(partials, blocks, out);
}